// PatchSampleF_84026740179061
// MI455X (gfx1250) — compile-verified
//
#include <hip/hip_runtime.h>
#include <hip/hip_bf16.h>
#include <math.h>

// ---------------------------------------------------------------------------
// PatchSampleF (CUT) for MI455X / gfx1250, wave32.
//   feats: (2,256,64,64) f32 x4 ; num_patches=256 ; out (4,512,256) f32
// Kernels:
//   k_sample_random : feats 0-2 gather + L2 norm (Feistel pseudo-permutation)
//   k_dots_entropy  : 7x7 local attention dots via V_WMMA_F32_16X16X4_F32
//                     (diagonal extraction); LDS staged with
//                     GLOBAL_LOAD_ASYNC_TO_LDS_B32 (ASYNCcnt path);
//                     attn written back with GLOBAL_STORE_ASYNC_FROM_LDS_B32
//   k_topk          : exact stable argsort (bitonic on u64 keys), take 256
//   k_attn_out      : attn-weighted neighbor sum + L2 norm
// ---------------------------------------------------------------------------

#define CCH   256
#define HW    64
#define LSZ   4096   /* 64*64 */
#define NPAT  256
#define KWIN  7
#define KK    49
#define BB    2

typedef float v2f __attribute__((ext_vector_type(2)));
typedef float v8f __attribute__((ext_vector_type(8)));

#if defined(__gfx1250__)
__device__ __forceinline__ void async_wait0() {
#if __has_builtin(__builtin_amdgcn_s_wait_asynccnt)
    __builtin_amdgcn_s_wait_asynccnt(0);
#else
    asm volatile("s_wait_asynccnt 0x0" ::: "memory");
#endif
}
#endif

// ------------------------- Feistel bijection on [0,4096) -------------------
// NOTE: stands in for jax.random.permutation (threefry not reproduced here).
__device__ __forceinline__ unsigned feistel12(unsigned x, unsigned seed) {
    unsigned l = x & 63u, r = (x >> 6) & 63u;
    #pragma unroll
    for (int i = 0; i < 4; ++i) {
        unsigned f = r * 2654435761u + seed * 0x9E3779B9u + (unsigned)i * 0x85EBCA6Bu;
        f ^= f >> 13; f *= 0xC2B2AE35u; f ^= f >> 11;
        unsigned nl = r;
        r = (l ^ f) & 63u;
        l = nl;
    }
    return (l << 6) | r;
}

// ------------------------- Kernel 1: random-sample feats 0..2 --------------
__global__ __launch_bounds__(256)
void k_sample_random(const float* __restrict__ f0,
                     const float* __restrict__ f1,
                     const float* __restrict__ f2,
                     float* __restrict__ out) {
    int blk     = blockIdx.x;               // 3 * B * NPAT blocks
    int feat_id = blk / (BB * NPAT);
    int rem     = blk % (BB * NPAT);
    int b       = rem / NPAT;
    int p       = rem % NPAT;
    const float* f = (feat_id == 0) ? f0 : (feat_id == 1) ? f1 : f2;

    unsigned loc = feistel12((unsigned)p, 0x51edu + (unsigned)feat_id * 0x9277u);

    int c = threadIdx.x;                    // one thread per channel
    float v = f[((size_t)(b * CCH + c)) * LSZ + loc];

    __shared__ float red[256];
    red[c] = v * v;
    __syncthreads();
    #pragma unroll
    for (int s = 128; s > 0; s >>= 1) {
        if (c < s) red[c] += red[c + s];
        __syncthreads();
    }
    float nrm = sqrtf(red[0]) + 1e-7f;
    out[(size_t)feat_id * (BB * NPAT * CCH) + (size_t)(b * NPAT + p) * CCH + c] = v / nrm;
}

// ------------------------- Kernel 2: dots + softmax + entropy --------------
// One block = 16 consecutive x at fixed y. 8 waves split the 49 offsets.
// dots[m, j] = sum_c F[c, y, x0+m] * F[c, y+dy_j, x0+m+dx_j]   (zero-padded)
__global__ __launch_bounds__(256)
void k_dots_entropy(const float* __restrict__ feat3,
                    float* __restrict__ attn_all,   // [B][LSZ][49]
                    float* __restrict__ entropy) {  // [B][LSZ]
    const int b   = blockIdx.x / (HW * (HW / 16));
    const int rem = blockIdx.x % (HW * (HW / 16));
    const int y0  = rem / (HW / 16);
    const int x0  = (rem % (HW / 16)) * 16;

    const int wave = threadIdx.x >> 5;      // 0..7
    const int lane = threadIdx.x & 31;

    // halo region: rows y0-3..y0+3, cols x0-3..x0+18, 64-channel chunk.
    // inner dim padded 64 -> 68 so WMMA fragment loads (col stride) hit
    // distinct LDS banks.
    __shared__ float reg[7][22][68];
    __shared__ float dots_tile[16][52];

#if defined(__gfx1250__)
    // Pre-zero the halo once: async loads below only overwrite in-bounds
    // slots, so zero-padding survives across all channel chunks.
    for (int e = threadIdx.x; e < 7 * 22 * 68; e += 256)
        (&reg[0][0][0])[e] = 0.f;
#endif

#if defined(__gfx1250__) && __has_builtin(__builtin_amdgcn_wmma_f32_16x16x4_f32)
    v8f acc[7];
    #pragma unroll
    for (int jj = 0; jj < 7; ++jj) {
        v8f z = {0.f,0.f,0.f,0.f,0.f,0.f,0.f,0.f};
        acc[jj] = z;
    }
#else
#warning "wmma_f32_16x16x4_f32 builtin unavailable: scalar fallback in k_dots_entropy"
    float sacc[7];
    #pragma unroll
    for (int jj = 0; jj < 7; ++jj) sacc[jj] = 0.f;
#endif

    for (int cc = 0; cc < CCH / 64; ++cc) {
        __syncthreads();                    // previous chunk fully consumed
        const int cbase = cc * 64;
        // cooperative halo load, 22-wide contiguous runs per (row, c)
        for (int e = threadIdx.x; e < 7 * 64 * 22; e += 256) {
            int col = e % 22;
            int rc  = e / 22;
            int ch  = rc % 64;
            int row = rc / 64;
            int yy = y0 - 3 + row;
            int xx = x0 - 3 + col;
#if defined(__gfx1250__)
            if ((unsigned)yy < (unsigned)HW && (unsigned)xx < (unsigned)HW) {
                // ASYNCcnt path: per-lane gather straight into LDS, no VGPR
                // round-trip. EXEC-masked (OOB) lanes keep their zeros.
                unsigned lds_off = (unsigned)(size_t)(&reg[row][col][ch]);
                unsigned long long gaddr = (unsigned long long)(size_t)
                    (feat3 + (((size_t)(b * CCH + cbase + ch)) * HW + yy) * HW + xx);
                asm volatile("global_load_async_to_lds_b32 %0, %1, off"
                             :: "v"(lds_off), "v"(gaddr) : "memory");
            }
#else
            float v = 0.f;
            if ((unsigned)yy < (unsigned)HW && (unsigned)xx < (unsigned)HW)
                v = feat3[(((size_t)(b * CCH + cbase + ch)) * HW + yy) * HW + xx];
            reg[row][col][ch] = v;
#endif
        }
#if defined(__gfx1250__)
        async_wait0();                      // my async->LDS writes landed
#endif
        __syncthreads();

        const int m    = lane & 15;         // location (A row) and B column
        const int koff = (lane >= 16) ? 2 : 0;

#if defined(__gfx1250__) && __has_builtin(__builtin_amdgcn_wmma_f32_16x16x4_f32)
        #pragma unroll
        for (int jj = 0; jj < 7; ++jj) {
            const int j = wave + jj * 8;    // wave-uniform -> EXEC stays full
            if (j < KK) {
                const int dy = j / KWIN - 3;
                const int dx = j % KWIN - 3;
                const float* arow = &reg[3][3 + m][0];
                const float* brow = &reg[3 + dy][3 + dx + m][0];
                #pragma unroll
                for (int ks = 0; ks < 16; ++ks) {
                    const int kb = ks * 4 + koff;
                    v2f afrag = *(const v2f*)(arow + kb);   // A: lanes<16 K=0,1 ; >=16 K=2,3
                    v2f bfrag = *(const v2f*)(brow + kb);   // B mirrored layout
                    acc[jj] = __builtin_amdgcn_wmma_f32_16x16x4_f32(
                        false, afrag, false, bfrag, (short)0, acc[jj], false, false);
                }
            }
        }
#else
        #pragma unroll
        for (int jj = 0; jj < 7; ++jj) {
            const int j = wave + jj * 8;
            if (j < KK && lane < 16) {
                const int dy = j / KWIN - 3;
                const int dx = j % KWIN - 3;
                float s = 0.f;
                for (int ch = 0; ch < 64; ++ch)
                    s += reg[3][3 + m][ch] * reg[3 + dy][3 + dx + m][ch];
                sacc[jj] += s;
            }
        }
#endif
    }

    // ---- write dots_tile ----
#if defined(__gfx1250__) && __has_builtin(__builtin_amdgcn_wmma_f32_16x16x4_f32)
    // D layout (16x16 f32): VGPR r, lanes 0-15 -> (M=r,   N=lane),
    //                               lanes16-31 -> (M=r+8, N=lane-16).
    // Diagonal n lives at: n<8 -> lane n, acc[n] ; n>=8 -> lane n+16, acc[n-8].
    {
        int dloc = -1;
        if (lane < 8)        dloc = lane;
        else if (lane >= 24) dloc = lane - 16;
        #pragma unroll
        for (int jj = 0; jj < 7; ++jj) {
            const int j = wave + jj * 8;
            if (j < KK && dloc >= 0) {
                float dv;
                switch (lane & 7) {
                    case 0: dv = acc[jj][0]; break;
                    case 1: dv = acc[jj][1]; break;
                    case 2: dv = acc[jj][2]; break;
                    case 3: dv = acc[jj][3]; break;
                    case 4: dv = acc[jj][4]; break;
                    case 5: dv = acc[jj][5]; break;
                    case 6: dv = acc[jj][6]; break;
                    default: dv = acc[jj][7]; break;
                }
                dots_tile[dloc][j] = dv;
            }
        }
    }
#else
    #pragma unroll
    for (int jj = 0; jj < 7; ++jj) {
        const int j = wave + jj * 8;
        if (j < KK && lane < 16) dots_tile[lane & 15][j] = sacc[jj];
    }
#endif
    __syncthreads();

    // ---- softmax + entropy over the 49 offsets, one thread per location ----
    // attn is produced in LDS (dots_tile overwritten in place), then
    // scattered to global via the async LDS->memory store path.
    if (threadIdx.x < 16) {
        const int m   = threadIdx.x;
        const int loc = y0 * HW + x0 + m;
        float mx = -INFINITY;
        for (int j = 0; j < KK; ++j) mx = fmaxf(mx, dots_tile[m][j]);
        float s = 0.f;
        for (int j = 0; j < KK; ++j) s += expf(dots_tile[m][j] - mx);
        const float inv = 1.f / s;
        float ent = 0.f;
        for (int j = 0; j < KK; ++j) {
            float a = expf(dots_tile[m][j] - mx) * inv;
            dots_tile[m][j] = a;            // in-place: dots -> attn
            if (a > 0.f) ent += a * (-logf(a));   // matches isinf(prob)->0 handling
        }
        entropy[b * LSZ + loc] = ent;
    }
    __syncthreads();

    // ---- write attn tile to global ----
    for (int e = threadIdx.x; e < 16 * KK; e += 256) {
        int m = e / KK, j = e % KK;
        const int loc = y0 * HW + x0 + m;
#if defined(__gfx1250__)
        unsigned lds_off = (unsigned)(size_t)(&dots_tile[m][j]);
        unsigned long long gaddr = (unsigned long long)(size_t)
            (attn_all + ((size_t)(b * LSZ + loc)) * KK + j);
        asm volatile("global_store_async_from_lds_b32 %0, %1, off"
                     :: "v"(gaddr), "v"(lds_off) : "memory");
#else
        attn_all[((size_t)(b * LSZ + loc)) * KK + j] = dots_tile[m][j];
#endif
    }
#if defined(__gfx1250__)
    async_wait0();                          // drain ASYNCcnt before retiring
#endif
}

// ------------------------- Kernel 3: exact stable top-256 ------------------
__global__ __launch_bounds__(1024)
void k_topk(const float* __restrict__ entropy, int* __restrict__ patch_id) {
    __shared__ unsigned long long keys[LSZ];
    const int b = blockIdx.x;
    for (int i = threadIdx.x; i < LSZ; i += blockDim.x) {
        unsigned eb = __float_as_uint(entropy[b * LSZ + i]);   // entropy >= 0
        keys[i] = ((unsigned long long)eb << 32) | (unsigned)i; // stable ties
    }
    __syncthreads();
    for (int size = 2; size <= LSZ; size <<= 1) {
        for (int stride = size >> 1; stride > 0; stride >>= 1) {
            for (int t = threadIdx.x; t < LSZ / 2; t += blockDim.x) {
                int i = ((t & ~(stride - 1)) << 1) | (t & (stride - 1));
                int p = i | stride;
                bool up = ((i & size) == 0);
                unsigned long long a = keys[i], c = keys[p];
                if ((a > c) == up) { keys[i] = c; keys[p] = a; }
            }
            __syncthreads();
        }
    }
    for (int i = threadIdx.x; i < NPAT; i += blockDim.x)
        patch_id[b * NPAT + i] = (int)(keys[i] & 0xFFFFFFFFull);
}

// ------------------------- Kernel 4: attn-weighted sum + L2 ----------------
__global__ __launch_bounds__(256)
void k_attn_out(const float* __restrict__ feat3,
                const float* __restrict__ attn_all,
                const int* __restrict__ patch_id,
                float* __restrict__ out) {
    const int blk = blockIdx.x;             // B * NPAT
    const int b = blk / NPAT, p = blk % NPAT;
    const int loc = patch_id[b * NPAT + p];
    const int y = loc >> 6, x = loc & 63;

    __shared__ float a_s[KK];
    __shared__ float red[256];
    const int c = threadIdx.x;
    if (c < KK) a_s[c] = attn_all[((size_t)(b * LSZ + loc)) * KK + c];
    __syncthreads();

    float acc = 0.f;
    const float* fb = feat3 + ((size_t)(b * CCH + c)) * LSZ;
    #pragma unroll
    for (int k = 0; k < KK; ++k) {
        int dy = k / KWIN - 3, dx = k % KWIN - 3;
        int yy = y + dy, xx = x + dx;
        if ((unsigned)yy < (unsigned)HW && (unsigned)xx < (unsigned)HW)
            acc += a_s[k] * fb[yy * HW + xx];
    }

    red[c] = acc * acc;
    __syncthreads();
    #pragma unroll
    for (int s = 128; s > 0; s >>= 1) {
        if (c < s) red[c] += red[c + s];
        __syncthreads();
    }
    float nrm = sqrtf(red[0]) + 1e-7f;
    out[(size_t)3 * (BB * NPAT * CCH) + (size_t)(b * NPAT + p) * CCH + c] = acc / nrm;
}

// ---------------------------------------------------------------------------
extern "C" void kernel_launch(void* const* d_in, const int* in_sizes, int n_in,
                              void* d_out, int out_size, void* d_ws, size_t ws_size,
                              hipStream_t stream) {
    const float* f0 = (const float*)d_in[0];
    const float* f1 = (const float*)d_in[1];
    const float* f2 = (const float*)d_in[2];
    const float* f3 = (const float*)d_in[3];
    float* out = (float*)d_out;

    // workspace layout
    float* attn_all = (float*)d_ws;                             // B*LSZ*49 f32
    float* entropy  = attn_all + (size_t)BB * LSZ * KK;         // B*LSZ f32
    int*   patch_id = (int*)(entropy + (size_t)BB * LSZ);       // B*NPAT i32

    (void)in_sizes; (void)n_in; (void)out_size; (void)ws_size;

    k_sample_random<<<3 * BB * NPAT, 256, 0, stream>>>(f0, f1, f2, out);
    k_dots_entropy<<<BB * HW * (HW / 16), 256, 0, stream>>>(f3, attn_all, entropy);
    k_topk<<<BB, 1024, 0, stream>>>(entropy, patch_id);
    k_attn_out<<<BB * NPAT, 256, 0, stream>>>(f3, attn_all, patch_id, out);
}